// SortPool_37039797961231
// MI455X (gfx1250) — compile-verified
//
#include <hip/hip_runtime.h>
#include <hip/hip_bf16.h>

typedef __attribute__((ext_vector_type(2))) float v2f;
typedef __attribute__((ext_vector_type(8))) float v8f;

#define N_NODES 50000
#define N_EDGES 800000
#define NFEAT 128
#define NHID 64
#define NB 512
#define KPOOL 30
#define NCONV 32
#define LCONV 26
#define NCLS 10

// ---------------------------------------------------------------------------
// utility kernels
// ---------------------------------------------------------------------------
__global__ void zero_f32(float* p, int n) {
    int i = blockIdx.x * blockDim.x + threadIdx.x;
    if (i < n) p[i] = 0.0f;
}
__global__ void zero_i32(int* p, int n) {
    int i = blockIdx.x * blockDim.x + threadIdx.x;
    if (i < n) p[i] = 0;
}

// ---------------------------------------------------------------------------
// fp32 WMMA GEMM: OUT[M,N] = A[M,K] @ W[K,N] (+bias) (+relu)
// M % 16 == 0, N % 16 == 0, K % 4 == 0. One wave -> one 16x16 tile,
// V_WMMA_F32_16X16X4_F32 accumulate over K.
// A 16x4 f32 layout: lanes 0-15 -> M=lane, v0=K0,v1=K1; lanes 16-31 -> K2,K3.
// B  4x16 f32 layout: lanes 0-15 -> N=lane, v0=K0,v1=K1; lanes 16-31 -> K2,K3.
// D 16x16 f32: vgpr r -> M = r + 8*half, N = lane%16.
// ---------------------------------------------------------------------------
__global__ void gemm_f32_wmma(const float* __restrict__ A, const float* __restrict__ W,
                              const float* __restrict__ bias, float* __restrict__ out,
                              int M, int K, int N, int fuse_relu) {
    int wave = threadIdx.x >> 5;
    int lane = threadIdx.x & 31;
    int tile = blockIdx.x * (blockDim.x >> 5) + wave;
    int tilesN = N >> 4;
    int tilesM = M >> 4;
    int tm = tile / tilesN;
    int tn = tile - tm * tilesN;
    if (tm >= tilesM) return;                 // wave-uniform: EXEC stays all-ones

    int half = lane >> 4;
    int lr   = lane & 15;
    const float* arow = A + (size_t)((tm << 4) + lr) * K;
    int col = (tn << 4) + lr;

    v8f acc = {};
    for (int k = 0; k < K; k += 4) {
        int ka = k + (half << 1);
        v2f a, b;
        a.x = arow[ka];
        a.y = arow[ka + 1];
        b.x = W[(size_t)ka * N + col];
        b.y = W[(size_t)(ka + 1) * N + col];
        acc = __builtin_amdgcn_wmma_f32_16x16x4_f32(false, a, false, b,
                                                    (short)0, acc, false, false);
    }

    float bv = bias ? bias[col] : 0.0f;
    int rowbase = (tm << 4) + (half << 3);
#pragma unroll
    for (int r = 0; r < 8; ++r) {
        float v = acc[r] + bv;
        if (fuse_relu) v = fmaxf(v, 0.0f);
        out[(size_t)(rowbase + r) * N + col] = v;
    }
}

// ---------------------------------------------------------------------------
// graph plumbing
// ---------------------------------------------------------------------------
__global__ void deg_kernel(const int* __restrict__ dst, float* __restrict__ deg, int nE) {
    int e = blockIdx.x * blockDim.x + threadIdx.x;
    if (e < nE) atomicAdd(&deg[dst[e]], 1.0f);
}

// agg[dst] += yl[src] over 64 channels; thread = (edge, channel)
__global__ void edge_scatter(const float* __restrict__ y, const int* __restrict__ src,
                             const int* __restrict__ dst, float* __restrict__ agg,
                             long long total) {
    long long idx = (long long)blockIdx.x * blockDim.x + threadIdx.x;
    if (idx >= total) return;
    int e = (int)(idx >> 6);
    int c = (int)(idx & 63);
    atomicAdd(&agg[(size_t)dst[e] * NHID + c], y[(size_t)src[e] * NHID + c]);
}

// h = relu(agg/max(deg,1) + yr)
__global__ void combine_kernel(const float* __restrict__ agg, const float* __restrict__ deg,
                               const float* __restrict__ yr, float* __restrict__ h, int total) {
    int i = blockIdx.x * blockDim.x + threadIdx.x;
    if (i >= total) return;
    float d = fmaxf(deg[i >> 6], 1.0f);
    h[i] = fmaxf(agg[i] / d + yr[i], 0.0f);
}

__global__ void count_kernel(const int* __restrict__ batch, int* __restrict__ counts, int n) {
    int i = blockIdx.x * blockDim.x + threadIdx.x;
    if (i < n) atomicAdd(&counts[batch[i]], 1);
}

__global__ void scan_kernel(const int* __restrict__ counts, int* __restrict__ starts, int nb) {
    if (blockIdx.x == 0 && threadIdx.x == 0) {
        int s = 0;
        for (int g = 0; g < nb; ++g) { starts[g] = s; s += counts[g]; }
    }
}

// ---------------------------------------------------------------------------
// sort-pool: one wave per graph; K threshold-selection passes over last channel,
// stable descending (ties -> lower node index first), matching stable lexsort.
// ---------------------------------------------------------------------------
__global__ void topk_kernel(const float* __restrict__ h, const int* __restrict__ starts,
                            const int* __restrict__ counts, int* __restrict__ topk) {
    int g = blockIdx.x;
    int lane = threadIdx.x;
    int s = starts[g], cnt = counts[g];
    float lastV = __builtin_inff();
    int lastI = -1;
    for (int k = 0; k < KPOOL; ++k) {
        float bv = -__builtin_inff();
        int bi = 0x7fffffff;
        if (k < cnt) {
            for (int j = lane; j < cnt; j += 32) {
                int node = s + j;
                float v = h[(size_t)node * NHID + (NHID - 1)];
                bool elig = (v < lastV) || (v == lastV && node > lastI);
                if (elig && (v > bv || (v == bv && node < bi))) { bv = v; bi = node; }
            }
        }
        for (int off = 16; off > 0; off >>= 1) {
            float ov = __shfl_xor(bv, off, 32);
            int   oi = __shfl_xor(bi, off, 32);
            if (ov > bv || (ov == bv && oi < bi)) { bv = ov; bi = oi; }
        }
        if (lane == 0) topk[g * KPOOL + k] = (k < cnt) ? bi : -1;
        lastV = bv; lastI = bi;
    }
}

// ---------------------------------------------------------------------------
// Conv1d(64 -> 32, kernel 5, valid) over pooled [NHID, KPOOL] tile, + relu.
// Output laid out [B, 32, 26] row-major so reshape(B, 832) is the identity.
// ---------------------------------------------------------------------------
__global__ void conv_kernel(const float* __restrict__ h, const int* __restrict__ topk,
                            const float* __restrict__ cw, const float* __restrict__ cb,
                            float* __restrict__ out) {
    __shared__ float p[NHID][KPOOL + 2];
    int g = blockIdx.x;
    int tid = threadIdx.x;   // 256
    for (int t = tid; t < NHID * (KPOOL + 2); t += 256) ((float*)p)[t] = 0.0f;
    __syncthreads();
    for (int idx = tid; idx < NHID * KPOOL; idx += 256) {
        int k = idx >> 6, c = idx & 63;         // coalesced over channel
        int node = topk[g * KPOOL + k];
        if (node >= 0) p[c][k] = h[(size_t)node * NHID + c];
    }
    __syncthreads();
    for (int oidx = tid; oidx < NCONV * LCONV; oidx += 256) {
        int o = oidx / LCONV, t = oidx - o * LCONV;
        float acc = cb[o];
        for (int i = 0; i < NHID; ++i) {
            const float* w = &cw[(o * NHID + i) * 5];
#pragma unroll
            for (int hh = 0; hh < 5; ++hh) acc += p[i][t + hh] * w[hh];
        }
        out[(size_t)g * (NCONV * LCONV) + oidx] = fmaxf(acc, 0.0f);
    }
}

// ---------------------------------------------------------------------------
// head: logits = f @ W2 + b2; log_softmax. One thread per row.
// ---------------------------------------------------------------------------
__global__ void head_kernel(const float* __restrict__ f, const float* __restrict__ w2,
                            const float* __restrict__ b2, float* __restrict__ out) {
    int row = blockIdx.x * blockDim.x + threadIdx.x;
    if (row >= NB) return;
    float lg[NCLS];
#pragma unroll
    for (int c = 0; c < NCLS; ++c) lg[c] = b2[c];
    for (int i = 0; i < NHID; ++i) {
        float x = f[(size_t)row * NHID + i];
#pragma unroll
        for (int c = 0; c < NCLS; ++c) lg[c] += x * w2[i * NCLS + c];
    }
    float m = lg[0];
#pragma unroll
    for (int c = 1; c < NCLS; ++c) m = fmaxf(m, lg[c]);
    float ss = 0.0f;
#pragma unroll
    for (int c = 0; c < NCLS; ++c) ss += expf(lg[c] - m);
    float lse = m + logf(ss);
#pragma unroll
    for (int c = 0; c < NCLS; ++c) out[row * NCLS + c] = lg[c] - lse;
}

// ---------------------------------------------------------------------------
// launcher
// ---------------------------------------------------------------------------
static inline int cdiv(long long a, int b) { return (int)((a + b - 1) / b); }

extern "C" void kernel_launch(void* const* d_in, const int* in_sizes, int n_in,
                              void* d_out, int out_size, void* d_ws, size_t ws_size,
                              hipStream_t stream) {
    const float* x     = (const float*)d_in[0];
    const int*   eidx  = (const int*)d_in[1];
    const int*   batch = (const int*)d_in[2];
    const float* W1l = (const float*)d_in[3];
    const float* b1  = (const float*)d_in[4];
    const float* W1r = (const float*)d_in[5];
    const float* W2l = (const float*)d_in[6];
    const float* b2  = (const float*)d_in[7];
    const float* W2r = (const float*)d_in[8];
    const float* W3l = (const float*)d_in[9];
    const float* b3  = (const float*)d_in[10];
    const float* W3r = (const float*)d_in[11];
    const float* convw = (const float*)d_in[12];
    const float* convb = (const float*)d_in[13];
    const float* lin1w = (const float*)d_in[14];
    const float* lin1b = (const float*)d_in[15];
    const float* lin2w = (const float*)d_in[16];
    const float* lin2b = (const float*)d_in[17];
    float* out = (float*)d_out;

    const int* src = eidx;
    const int* dst = eidx + N_EDGES;

    // workspace carve-up
    char* ws = (char*)d_ws;
    const size_t featB = (size_t)N_NODES * NHID * sizeof(float);
    float* hA  = (float*)ws;                 ws += featB;
    float* hB  = (float*)ws;                 ws += featB;
    float* yl  = (float*)ws;                 ws += featB;
    float* yr  = (float*)ws;                 ws += featB;
    float* agg = (float*)ws;                 ws += featB;
    float* deg = (float*)ws;                 ws += (size_t)N_NODES * sizeof(float);
    int* counts = (int*)ws;                  ws += NB * sizeof(int);
    int* starts = (int*)ws;                  ws += NB * sizeof(int);
    int* topk   = (int*)ws;                  ws += (size_t)NB * KPOOL * sizeof(int);
    float* convout = yl;                     // reuse after layer 3
    float* f1      = yr;                     // reuse after layer 3

    const int ZB = 256;
    const int featTotal = N_NODES * NHID;
    const long long scatTotal = (long long)N_EDGES * NHID;

    // degrees + per-graph node counts/starts (edges/batch are layer-invariant)
    zero_f32<<<cdiv(N_NODES, ZB), ZB, 0, stream>>>(deg, N_NODES);
    zero_i32<<<cdiv(NB, ZB), ZB, 0, stream>>>(counts, NB);
    deg_kernel<<<cdiv(N_EDGES, ZB), ZB, 0, stream>>>(dst, deg, N_EDGES);
    count_kernel<<<cdiv(N_NODES, ZB), ZB, 0, stream>>>(batch, counts, N_NODES);
    scan_kernel<<<1, 32, 0, stream>>>(counts, starts, NB);

    // gemm launch helper dims: 4 waves (tiles) per 128-thread block
    auto gemm_blocks = [](int M, int N) { return cdiv((M >> 4) * (N >> 4), 4); };

    // ---- SAGE layer: h_out = relu( mean_agg(h_in @ Wl) + (h_in @ Wr + b) )
    // (mean commutes with the linear map, so aggregate AFTER the GEMM: 64-ch edges)
    struct Layer { const float* in; int K; const float* Wl; const float* b;
                   const float* Wr; float* outp; };
    Layer layers[3] = {
        { x,  NFEAT, W1l, b1, W1r, hA },
        { hA, NHID,  W2l, b2, W2r, hB },
        { hB, NHID,  W3l, b3, W3r, hA },
    };
    for (int L = 0; L < 3; ++L) {
        zero_f32<<<cdiv(featTotal, ZB), ZB, 0, stream>>>(agg, featTotal);
        gemm_f32_wmma<<<gemm_blocks(N_NODES, NHID), 128, 0, stream>>>(
            layers[L].in, layers[L].Wl, nullptr, yl, N_NODES, layers[L].K, NHID, 0);
        gemm_f32_wmma<<<gemm_blocks(N_NODES, NHID), 128, 0, stream>>>(
            layers[L].in, layers[L].Wr, layers[L].b, yr, N_NODES, layers[L].K, NHID, 0);
        edge_scatter<<<cdiv(scatTotal, ZB), ZB, 0, stream>>>(yl, src, dst, agg, scatTotal);
        combine_kernel<<<cdiv(featTotal, ZB), ZB, 0, stream>>>(agg, deg, yr,
                                                               layers[L].outp, featTotal);
    }

    // ---- sort-pool top-K per graph (stable descending by last channel of hA)
    topk_kernel<<<NB, 32, 0, stream>>>(hA, starts, counts, topk);

    // ---- Conv1d + relu -> [B, 32, 26]
    conv_kernel<<<NB, 256, 0, stream>>>(hA, topk, convw, convb, convout);

    // ---- lin1 (832 -> 64) + relu, via WMMA
    gemm_f32_wmma<<<gemm_blocks(NB, NHID), 128, 0, stream>>>(
        convout, lin1w, lin1b, f1, NB, NCONV * LCONV, NHID, 1);

    // ---- lin2 (64 -> 10) + log_softmax
    head_kernel<<<cdiv(NB, 64), 64, 0, stream>>>(f1, lin2w, lin2b, out);
}